// GAT_32908039422460
// MI455X (gfx1250) — compile-verified
//
#include <hip/hip_runtime.h>

typedef __attribute__((ext_vector_type(2))) float v2f;
typedef __attribute__((ext_vector_type(8))) float v8f;

#define NEG_SLOPE 0.2f
#define EPSV 1e-16f

// order-preserving float -> uint encoding for atomicMax-based segment max
__device__ __forceinline__ unsigned enc_ord(float f) {
  unsigned u = __float_as_uint(f);
  return (u & 0x80000000u) ? ~u : (u | 0x80000000u);
}
__device__ __forceinline__ float dec_ord(unsigned u) {
  unsigned v = (u & 0x80000000u) ? (u & 0x7FFFFFFFu) : ~u;
  return __uint_as_float(v);
}

// C[M,Nout] = A[M,K] x B[K,Nout], fp32, one 16x16 tile per wave via
// V_WMMA_F32_16X16X4_F32 (exact fp32 matrix op; K stepped by 4).
__global__ __launch_bounds__(256) void gat_gemm_wmma(
    const float* __restrict__ A, const float* __restrict__ B,
    float* __restrict__ C, int M, int K, int Nout)
{
  const int wave = (int)((blockIdx.x * blockDim.x + threadIdx.x) >> 5);
  const int lane = (int)(threadIdx.x & 31);
  const int colTiles = Nout >> 4;
  const int rowTile = wave / colTiles;
  const int colTile = wave - rowTile * colTiles;
  if (rowTile * 16 >= M) return;            // uniform per wave -> EXEC all-1 below
  const int lo = lane & 15;                 // M (for A) / N (for B,C)
  const int hi = lane >> 4;                 // selects K pair 0..1 vs 2..3
  const float* aptr = A + (size_t)(rowTile * 16 + lo) * K + hi * 2;
  const float* bptr = B + (size_t)(hi * 2) * Nout + colTile * 16 + lo;
  v8f acc = {};
  for (int kk = 0; kk < K; kk += 4) {
    // A 16x4 frag: lanes0-15 K=kk,kk+1 ; lanes16-31 K=kk+2,kk+3 (8B aligned)
    v2f a = *(const v2f*)(aptr + kk);
    v2f b;
    b.x = bptr[(size_t)kk * Nout];
    b.y = bptr[(size_t)(kk + 1) * Nout];
    acc = __builtin_amdgcn_wmma_f32_16x16x4_f32(
        /*neg_a=*/false, a, /*neg_b=*/false, b,
        /*c_mod=*/(short)0, acc, /*reuse_a=*/false, /*reuse_b=*/false);
  }
  // D layout: VGPR r -> M = rowTile*16 + r + 8*hi, N = colTile*16 + lo
  float* cptr = C + (size_t)(rowTile * 16 + hi * 8) * Nout + colTile * 16 + lo;
#pragma unroll
  for (int r = 0; r < 8; ++r) cptr[(size_t)r * Nout] = acc[r];
}

// a_src[n,h] = h[n,h,:] . att_src[h,:]  (and same for dst); C must be mult of 4
template<int H, int C>
__global__ __launch_bounds__(256) void gat_att(
    const float* __restrict__ h, const float* __restrict__ att_src,
    const float* __restrict__ att_dst, float* __restrict__ a_src,
    float* __restrict__ a_dst, int n)
{
  int idx = blockIdx.x * blockDim.x + threadIdx.x;
  if (idx >= n * H) return;
  int node = idx / H;
  int hh = idx - node * H;
  const float4* hv = (const float4*)(h + ((size_t)node * H + hh) * C);
  const float4* as = (const float4*)(att_src + hh * C);
  const float4* ad = (const float4*)(att_dst + hh * C);
  float ssum = 0.f, dsum = 0.f;
#pragma unroll
  for (int q = 0; q < C / 4; ++q) {
    float4 v = hv[q];
    float4 s = as[q];
    float4 d = ad[q];
    ssum += v.x * s.x + v.y * s.y + v.z * s.z + v.w * s.w;
    dsum += v.x * d.x + v.y * d.y + v.z * d.z + v.w * d.w;
  }
  a_src[idx] = ssum;
  a_dst[idx] = dsum;
}

// segment max of leaky_relu(a_src[src]+a_dst[dst]) over dst via encoded atomicMax
template<int H>
__global__ __launch_bounds__(256) void gat_edge_max(
    const long long* __restrict__ ei, int Eraw, int Etot,
    const float* __restrict__ a_src, const float* __restrict__ a_dst,
    unsigned* __restrict__ amax)
{
  int idx = blockIdx.x * blockDim.x + threadIdx.x;
  if (idx >= Etot * H) return;
  int e = idx / H;
  int hh = idx - e * H;
  int s, d;
  if (e < Eraw) { s = (int)ei[e]; d = (int)ei[Eraw + e]; }
  else          { s = e - Eraw;   d = s; }
  float al = a_src[s * H + hh] + a_dst[d * H + hh];
  al = al > 0.0f ? al : NEG_SLOPE * al;
  atomicMax(amax + (d * H + hh), enc_ord(al));
}

// fused: denom[dst,h] += ex ; agg[dst,h,:] += ex * feat[src,h,:]
// (coef = ex/denom is linear in the aggregation -> normalize once per node later)
template<int H, int C>
__global__ __launch_bounds__(256) void gat_edge_accum(
    const long long* __restrict__ ei, int Eraw, int Etot,
    const float* __restrict__ a_src, const float* __restrict__ a_dst,
    const unsigned* __restrict__ amax, const float* __restrict__ feat,
    float* __restrict__ denom, float* __restrict__ agg)
{
  int idx = blockIdx.x * blockDim.x + threadIdx.x;
  if (idx >= Etot * H) return;
  int e = idx / H;
  int hh = idx - e * H;
  int s, d;
  if (e < Eraw) { s = (int)ei[e]; d = (int)ei[Eraw + e]; }
  else          { s = e - Eraw;   d = s; }
  float al = a_src[s * H + hh] + a_dst[d * H + hh];
  al = al > 0.0f ? al : NEG_SLOPE * al;
  float ex = __expf(al - dec_ord(amax[d * H + hh]));
  atomicAdd(denom + (d * H + hh), ex);
  const float4* fv = (const float4*)(feat + ((size_t)s * H + hh) * C);
  float* av = agg + ((size_t)d * H + hh) * C;
#pragma unroll
  for (int q = 0; q < C / 4; ++q) {
    float4 v = fv[q];                        // wide b128 gather
    atomicAdd(av + 4 * q + 0, ex * v.x);
    atomicAdd(av + 4 * q + 1, ex * v.y);
    atomicAdd(av + 4 * q + 2, ex * v.z);
    atomicAdd(av + 4 * q + 3, ex * v.w);
  }
}

// out[n,h,:4q] = agg/(denom[n,h]+eps) + bias  (optional ELU); vectorized x4
template<int H, int C>
__global__ __launch_bounds__(256) void gat_normalize(
    const float* __restrict__ agg, const float* __restrict__ denom,
    const float* __restrict__ bias, float* __restrict__ outp, int n, int apply_elu)
{
  int idx = blockIdx.x * blockDim.x + threadIdx.x;   // one float4 per thread
  int total = n * H * (C / 4);
  if (idx >= total) return;
  int node = idx / (H * (C / 4));
  int r = idx - node * (H * (C / 4));                // packed (h, c/4)
  int hh = r / (C / 4);
  float inv = 1.0f / (denom[node * H + hh] + EPSV);
  float4 v = ((const float4*)agg)[idx];
  const float4 b = ((const float4*)bias)[r];
  v.x = v.x * inv + b.x;
  v.y = v.y * inv + b.y;
  v.z = v.z * inv + b.z;
  v.w = v.w * inv + b.w;
  if (apply_elu) {
    v.x = v.x > 0.0f ? v.x : (__expf(v.x) - 1.0f);
    v.y = v.y > 0.0f ? v.y : (__expf(v.y) - 1.0f);
    v.z = v.z > 0.0f ? v.z : (__expf(v.z) - 1.0f);
    v.w = v.w > 0.0f ? v.w : (__expf(v.w) - 1.0f);
  }
  ((float4*)outp)[idx] = v;
}

extern "C" void kernel_launch(void* const* d_in, const int* in_sizes, int n_in,
                              void* d_out, int out_size, void* d_ws, size_t ws_size,
                              hipStream_t stream)
{
  (void)n_in; (void)out_size; (void)ws_size;
  const float*     x   = (const float*)d_in[0];
  const long long* ei  = (const long long*)d_in[1];   // int64 [2,E]
  const float*     W1  = (const float*)d_in[2];
  const float*     as1 = (const float*)d_in[3];
  const float*     ad1 = (const float*)d_in[4];
  const float*     b1  = (const float*)d_in[5];
  const float*     W2  = (const float*)d_in[6];
  const float*     as2 = (const float*)d_in[7];
  const float*     ad2 = (const float*)d_in[8];
  const float*     b2  = (const float*)d_in[9];
  float* out = (float*)d_out;

  const int NN = 100000, FEAT = 128, HC1 = 128;
  const int Eraw = in_sizes[1] / 2;
  const int Etot = Eraw + NN;

  char* ws = (char*)d_ws;
  float*    h1    = (float*)(ws);                 // N*128  (reused as h2 after conv1)
  float*    agg1  = (float*)(ws + 51200000);      // N*128
  float*    asrc1 = (float*)(ws + 102400000);     // N*8
  float*    adst1 = (float*)(ws + 105600000);     // N*8
  unsigned* amax1 = (unsigned*)(ws + 108800000);  // N*8
  float*    den1  = (float*)(ws + 112000000);     // N*8
  float*    g2    = (float*)(ws + 115200000);     // N*16
  float*    asrc2 = (float*)(ws + 121600000);     // N
  float*    adst2 = (float*)(ws + 122000000);     // N
  unsigned* amax2 = (unsigned*)(ws + 122400000);  // N
  float*    den2  = (float*)(ws + 122800000);     // N

  hipMemsetAsync(agg1,  0, 51200000, stream);               // agg1
  hipMemsetAsync(amax1, 0, 6400000,  stream);               // amax1 + den1
  hipMemsetAsync(amax2, 0, 800000,   stream);               // amax2 + den2
  hipMemsetAsync(out,   0, (size_t)NN * 16 * 4, stream);    // agg2 == d_out

  // ---- conv1 ----
  {
    int waves  = (NN / 16) * (HC1 / 16);
    int blocks = (waves * 32 + 255) / 256;
    gat_gemm_wmma<<<blocks, 256, 0, stream>>>(x, W1, h1, NN, FEAT, HC1);
  }
  gat_att<8, 16><<<(NN * 8 + 255) / 256, 256, 0, stream>>>(h1, as1, ad1, asrc1, adst1, NN);
  gat_edge_max<8><<<(Etot * 8 + 255) / 256, 256, 0, stream>>>(ei, Eraw, Etot, asrc1, adst1, amax1);
  gat_edge_accum<8, 16><<<(Etot * 8 + 255) / 256, 256, 0, stream>>>(
      ei, Eraw, Etot, asrc1, adst1, amax1, h1, den1, agg1);
  gat_normalize<8, 16><<<(NN * 32 + 255) / 256, 256, 0, stream>>>(
      agg1, den1, b1, h1, NN, /*elu=*/1);    // h2 written over h1

  // ---- conv2 ----
  {
    int waves  = (NN / 16) * (16 / 16);
    int blocks = (waves * 32 + 255) / 256;
    gat_gemm_wmma<<<blocks, 256, 0, stream>>>(h1, W2, g2, NN, HC1, 16);
  }
  gat_att<1, 16><<<(NN + 255) / 256, 256, 0, stream>>>(g2, as2, ad2, asrc2, adst2, NN);
  gat_edge_max<1><<<(Etot + 255) / 256, 256, 0, stream>>>(ei, Eraw, Etot, asrc2, adst2, amax2);
  gat_edge_accum<1, 16><<<(Etot + 255) / 256, 256, 0, stream>>>(
      ei, Eraw, Etot, asrc2, adst2, amax2, g2, den2, out);
  gat_normalize<1, 16><<<(NN * 4 + 255) / 256, 256, 0, stream>>>(
      out, den2, b2, out, NN, /*elu=*/0);
}